// MaxUnpooling2D_62259845923142
// MI455X (gfx1250) — compile-verified
//
#include <hip/hip_runtime.h>

// MaxUnpooling2D scatter-add for MI455X (gfx1250, wave32).
//
// updates : (B=32, H=64, W=64, C=128) float32, contiguous  -> 16,777,216 elems
// mask    : same shape int32, values in [0, 2^21)          -> per-batch flat index
// out     : (B=32, 128, 128, 128) float32                  -> 67,108,864 elems
//
// Pass 1: zero-fill output with 128-bit stores.
// Pass 2: each thread loads 4 updates + 4 indices (b128, non-temporal) and
//         issues 4 no-return agent-scope global_atomic_add_f32 into its
//         batch's 8 MB output slab (L2-resident given batch-ordered blocks).

typedef __attribute__((ext_vector_type(4))) float v4f;
typedef __attribute__((ext_vector_type(4))) int   v4i;

constexpr int IN_PER_BATCH_LOG2 = 19;                    // 64*64*128  = 2^19
constexpr int IN_PER_BATCH      = 1 << IN_PER_BATCH_LOG2;
constexpr int OUT_PER_BATCH     = 128 * 128 * 128;       // 2^21
static_assert(IN_PER_BATCH == 64 * 64 * 128, "batch stride mismatch");

__global__ __launch_bounds__(256) void zero_fill_b128(v4f* __restrict__ out, int n4) {
    int i = blockIdx.x * 256 + threadIdx.x;
    if (i < n4) {
        v4f z = {0.0f, 0.0f, 0.0f, 0.0f};
        // Plain RT store -> global_store_b128; dirty-zero lines flow through L2.
        out[i] = z;
    }
}

__global__ __launch_bounds__(256) void scatter_add_b128(const float* __restrict__ upd,
                                                        const int* __restrict__ mask,
                                                        float* __restrict__ out,
                                                        int n) {
    int i = (blockIdx.x * 256 + threadIdx.x) * 4;
    if (i >= n) return;

    // Streamed exactly once -> non-temporal b128 loads (don't pollute L2,
    // which we want full of atomic-target output lines).
    v4f u = __builtin_nontemporal_load(reinterpret_cast<const v4f*>(upd + i));
    v4i m = __builtin_nontemporal_load(reinterpret_cast<const v4i*>(mask + i));

    // All 4 elements share one batch (i is a multiple of 4, batch stride 2^19).
    float* obase = out + (size_t)(i >> IN_PER_BATCH_LOG2) * (size_t)OUT_PER_BATCH;

    // No-return, relaxed, agent-scope float adds -> global_atomic_add_f32
    // (hardware L2 RMW; duplicates in mask sum correctly, matching scatter_nd).
    __hip_atomic_fetch_add(obase + m.x, u.x, __ATOMIC_RELAXED, __HIP_MEMORY_SCOPE_AGENT);
    __hip_atomic_fetch_add(obase + m.y, u.y, __ATOMIC_RELAXED, __HIP_MEMORY_SCOPE_AGENT);
    __hip_atomic_fetch_add(obase + m.z, u.z, __ATOMIC_RELAXED, __HIP_MEMORY_SCOPE_AGENT);
    __hip_atomic_fetch_add(obase + m.w, u.w, __ATOMIC_RELAXED, __HIP_MEMORY_SCOPE_AGENT);
}

extern "C" void kernel_launch(void* const* d_in, const int* in_sizes, int n_in,
                              void* d_out, int out_size, void* d_ws, size_t ws_size,
                              hipStream_t stream) {
    const float* upd  = (const float*)d_in[0];
    const int*   mask = (const int*)d_in[1];
    float*       out  = (float*)d_out;
    const int n = in_sizes[0];          // 16,777,216 update elements

    // Pass 1: zero the (poisoned) output. out_size = 67,108,864 (divisible by 4).
    {
        int n4 = out_size >> 2;
        int blocks = (n4 + 255) / 256;  // 65,536 blocks x 8 waves
        zero_fill_b128<<<blocks, 256, 0, stream>>>((v4f*)out, n4);
    }

    // Pass 2: scatter-add, 4 elements per thread.
    {
        int n4 = (n + 3) >> 2;
        int blocks = (n4 + 255) / 256;  // 16,384 blocks
        scatter_add_b128<<<blocks, 256, 0, stream>>>(upd, mask, out, n);
    }
}